// AnchorFreeLoss_63453846831417
// MI455X (gfx1250) — compile-verified
//
#include <hip/hip_runtime.h>
#include <hip/hip_bf16.h>

// ---------------- problem constants (from reference) ----------------
#define B_      16
#define M_      64
#define H_      160
#define W_      160
#define HW_     (H_ * W_)
#define C_      43
#define STRIDE_ 4.0f          // IMG_SIZE / H = 640 / 160
#define EPS_    1e-7f

typedef __attribute__((ext_vector_type(2))) float v2f;
typedef __attribute__((ext_vector_type(8))) float v8f;
typedef __attribute__((ext_vector_type(4))) unsigned int u32x4;
typedef __attribute__((ext_vector_type(4))) int          i32x4;
typedef __attribute__((ext_vector_type(8))) int          i32x8;

// ---------------------------------------------------------------
// Wave32 sum reduction using V_WMMA_F32_16X16X4_F32.
// A = ones(16x4), B[k][n] holds per-lane partials (b0 = v, b1 = 0)
// => D[m][n] = sum_k B[k][n] = v[lane n] + v[lane n+16], replicated
// over all rows m. D vgpr0 at lane L = colsum[L % 16]; finish with
// a 4-step xor-shuffle over the 16 columns. Requires full EXEC
// (call sites are unconditional in a 256-thread block).
// ---------------------------------------------------------------
__device__ __forceinline__ float wave_sum_wmma(float v) {
    v2f a; a[0] = 1.0f; a[1] = 1.0f;
    v2f b; b[0] = v;    b[1] = 0.0f;
    v8f c = {};
    c = __builtin_amdgcn_wmma_f32_16x16x4_f32(
            /*neg_a=*/false, a, /*neg_b=*/false, b,
            /*c_mod=*/(short)0, c, /*reuse_a=*/false, /*reuse_b=*/false);
    float s = c[0];
    s += __shfl_xor(s, 8, 32);
    s += __shfl_xor(s, 4, 32);
    s += __shfl_xor(s, 2, 32);
    s += __shfl_xor(s, 1, 32);
    return s;
}

// ---------------- stage 0: zero class-bitmask + accumulators ----------------
__global__ void afl_zero_kernel(unsigned int* clsbits32, float* acc) {
    int i = blockIdx.x * blockDim.x + threadIdx.x;
    if (i < 8) acc[i] = 0.0f;
    if (i < B_ * HW_ * 2) clsbits32[i] = 0u;
}

// ---------------- stage 1: per-object targets + scatter ----------------
// Writes per-object gaussian params (gx, gy, r; r<0 => invalid), ORs the
// class bit into the 64-bit per-cell bitmask (dedups mask + class target),
// and accumulates the per-object box L1 directly (no dense box target).
__global__ void afl_obj_kernel(const float* __restrict__ bboxes,
                               const int*   __restrict__ labels,
                               const float* __restrict__ pred_boxes,
                               float* __restrict__ objx,
                               float* __restrict__ objy,
                               float* __restrict__ objr,
                               unsigned long long* __restrict__ clsbits,
                               float* __restrict__ acc) {
    int i = blockIdx.x * blockDim.x + threadIdx.x;
    if (i >= B_ * M_) return;
    int b = i / M_;

    float x1 = bboxes[i * 4 + 0];
    float y1 = bboxes[i * 4 + 1];
    float x2 = bboxes[i * 4 + 2];
    float y2 = bboxes[i * 4 + 3];
    int   lab_raw = labels[i];

    float cx = 0.5f * (x1 + x2);
    float cy = 0.5f * (y1 + y2);
    float bw = x2 - x1;
    float bh = y2 - y1;
    bool valid = (lab_raw >= 0) && ((x1 + y1 + x2 + y2) > 0.0f) &&
                 (bw > 0.0f) && (bh > 0.0f);

    int gx = (int)(cx / STRIDE_);
    gx = gx < 0 ? 0 : (gx > W_ - 1 ? W_ - 1 : gx);
    int gy = (int)(cy / STRIDE_);
    gy = gy < 0 ? 0 : (gy > H_ - 1 ? H_ - 1 : gy);

    float rout = -1.0f;
    if (valid) {
        float rf = sqrtf(bw * bh) / STRIDE_;
        if (rf < 2.0f) rf = 2.0f;
        rout = (float)(int)rf;       // matches astype(int32).astype(float32)
    }
    objx[i] = (float)gx;
    objy[i] = (float)gy;
    objr[i] = rout;

    if (valid) {
        int lab = lab_raw;
        lab = lab < 0 ? 0 : (lab > C_ - 1 ? C_ - 1 : lab);
        int cell = b * HW_ + gy * W_ + gx;
        unsigned int* w = (unsigned int*)&clsbits[cell];
        if (lab < 32) atomicOr(&w[0], 1u << lab);
        else          atomicOr(&w[1], 1u << (lab - 32));

        float dx = (cx - ((float)gx + 0.5f) * STRIDE_) / STRIDE_;
        float dy = (cy - ((float)gy + 0.5f) * STRIDE_) / STRIDE_;
        float dw = __logf(bw / STRIDE_ + 1e-6f);
        float dh = __logf(bh / STRIDE_ + 1e-6f);

        const float* pb = pred_boxes + (size_t)b * 4 * HW_ + gy * W_ + gx;
        float l1 = fabsf(pb[0 * HW_] - dx) + fabsf(pb[1 * HW_] - dy) +
                   fabsf(pb[2 * HW_] - dw) + fabsf(pb[3 * HW_] - dh);
        atomicAdd(&acc[2], l1);
    }
}

// ---------------- stage 2: fused heatmap + focal + class losses ----------------
// 256 cells per block, 100 blocks per batch image. Object params staged into LDS
// by the Tensor Data Mover (2D tile: 3 rows x 64 floats, row stride 1024 floats),
// tracked via TENSORcnt. Only reads pred_classes at masked cells (~1k cells),
// avoiding 70 MB of traffic; the unmasked constant contribution is added
// analytically in the finalize stage.
__global__ void afl_main_kernel(const float* __restrict__ pred_heat,
                                const float* __restrict__ pred_cls,
                                const float* __restrict__ objx,   // rows: objx, objy, objr
                                const unsigned long long* __restrict__ clsbits,
                                float* __restrict__ acc) {
    __shared__ float s_par[3 * M_];   // [0..63]=gx, [64..127]=gy, [128..191]=r

    const int blocks_per_img = HW_ / 256;   // 100
    int b    = blockIdx.x / blocks_per_img;
    int base = (blockIdx.x % blocks_per_img) * 256;
    int t    = threadIdx.x;

    // ---- TDM: DMA the 3x64 object-parameter tile from global into LDS ----
    // D# group0: count=1, lds_addr, global_addr[56:0], type=2 ("image")
    // D# group1: data_size=2 (4B), tensor_dim0=64, tensor_dim1=3,
    //            tile_dim0=64, tile_dim1=3, tensor_dim0_stride=1024 elems
    if (t < 32) {   // wave 0 issues the DMA (TDM ignores EXEC; one wave = one op)
        unsigned lds_off = (unsigned)(size_t)(void*)s_par;
        unsigned long long ga =
            (unsigned long long)(size_t)(const void*)(objx + (size_t)b * M_);
        u32x4 g0;
        g0[0] = 1u;                                   // count = 1 valid descriptor
        g0[1] = lds_off;                              // lds_addr
        g0[2] = (unsigned)ga;                         // global_addr[31:0]
        g0[3] = ((unsigned)(ga >> 32) & 0x01FFFFFFu)  // global_addr[56:32]
                | (2u << 30);                         // type = 2
        i32x8 g1;
        g1[0] = (int)0x00020000u;       // workgroup_mask=0, data_size=2 (4 bytes)
        g1[1] = (int)(64u << 16);       // tensor_dim0[15:0] = 64
        g1[2] = (int)(3u << 16);        // tensor_dim0[31:16]=0, tensor_dim1[15:0]=3
        g1[3] = (int)(64u << 16);       // tensor_dim1[31:16]=0, tile_dim0 = 64
        g1[4] = 3;                      // tile_dim1 = 3, tile_dim2 = 0 (2D)
        g1[5] = 1024;                   // tensor_dim0_stride[31:0] = 1024 elems
        g1[6] = 0;                      // stride hi / tensor_dim1_stride lo
        g1[7] = 0;
        i32x4 gz; gz[0] = 0; gz[1] = 0; gz[2] = 0; gz[3] = 0;
#if defined(__clang_major__) && __clang_major__ >= 23
        i32x8 gz8;
        gz8[0] = 0; gz8[1] = 0; gz8[2] = 0; gz8[3] = 0;
        gz8[4] = 0; gz8[5] = 0; gz8[6] = 0; gz8[7] = 0;
        __builtin_amdgcn_tensor_load_to_lds(g0, g1, gz, gz, gz8, 0);
#else
        __builtin_amdgcn_tensor_load_to_lds(g0, g1, gz, gz, 0);
#endif
        __builtin_amdgcn_s_wait_tensorcnt(0);
    }
    __syncthreads();

    int cell = base + t;
    int y = cell / W_;
    int x = cell - y * W_;
    float fx = (float)x, fy = (float)y;

    size_t gi = (size_t)b * HW_ + cell;
    // CDNA5 prefetch of the first class-channel line for this cell
    __builtin_prefetch(pred_cls + (size_t)b * C_ * HW_ + cell, 0, 0);

    // ---- heatmap target: max over objects of clipped gaussian ----
    float tmax = 0.0f;
#pragma unroll 4
    for (int m = 0; m < M_; ++m) {
        float r   = s_par[128 + m];
        float dxx = fx - s_par[m];
        float dyy = fy - s_par[64 + m];
        float dsq = dxx * dxx + dyy * dyy;
        float r2  = r * r;
        if (r > 0.0f && dsq <= 4.0f * r2) {
            float g = __expf(-2.0f * dsq / r2);   // sigma = r/2 -> 2*s^2 = r^2/2
            tmax = fmaxf(tmax, g);
        }
    }

    // ---- heatmap focal loss ----
    float p = pred_heat[gi];
    p = fminf(fmaxf(p, EPS_), 1.0f - EPS_);
    float hsum, hpos;
    if (tmax > 0.5f) {
        hsum = -0.25f * (1.0f - p) * (1.0f - p) * __logf(p) * tmax;
        hpos = 1.0f;
    } else {
        hsum = -0.75f * p * p * __logf(1.0f - p) * (1.0f - tmax);
        hpos = 0.0f;
    }

    // ---- class focal loss (masked cells only) ----
    unsigned long long bits = clsbits[gi];
    float msk = 0.0f, csum = 0.0f, cpos = 0.0f;
    if (bits) {
        msk  = 1.0f;
        cpos = (float)__popcll(bits);
        const float* pc = pred_cls + (size_t)b * C_ * HW_ + cell;
#pragma unroll 1
        for (int c = 0; c < C_; ++c) {
            float z  = pc[(size_t)c * HW_];
            float s  = 1.0f / (1.0f + __expf(-z));
            float pp = fminf(fmaxf(s, EPS_), 1.0f - EPS_);
            if ((bits >> c) & 1ull)
                csum += -0.25f * (1.0f - pp) * (1.0f - pp) * __logf(pp);
            else
                csum += -0.75f * pp * pp * __logf(1.0f - pp);
        }
    }

    // ---- on-chip reduction: WMMA wave-sum + cross-wave atomics ----
    float w0 = wave_sum_wmma(hsum);
    float w1 = wave_sum_wmma(hpos);
    float w2 = wave_sum_wmma(msk);
    float w3 = wave_sum_wmma(csum);
    float w4 = wave_sum_wmma(cpos);
    if ((t & 31) == 0) {
        atomicAdd(&acc[0], w0);
        atomicAdd(&acc[1], w1);
        atomicAdd(&acc[3], w2);
        atomicAdd(&acc[4], w3);
        atomicAdd(&acc[5], w4);
    }
}

// ---------------- stage 3: finalize scalar ----------------
__global__ void afl_final_kernel(const float* __restrict__ acc,
                                 float* __restrict__ out) {
    float heat_sum = acc[0];
    float heat_pos = acc[1];
    float l1       = acc[2];
    float maskc    = acc[3];
    float cls_sum  = acc[4];
    float cls_pos  = acc[5];

    float heat_loss = heat_sum / fmaxf(heat_pos, 1.0f);

    float num_pos  = fmaxf(maskc, 1.0f);
    float box_loss = (num_pos > 1.0f) ? (l1 / num_pos) : 0.0f;

    // analytic constant for all unmasked (cell,channel) elements:
    // p = eps, target = 0  ->  -(1-alpha)*eps^2*log(1-eps)
    float c0 = -0.75f * EPS_ * EPS_ * __logf(1.0f - EPS_);
    float n_unmasked = ((float)(B_ * HW_) - maskc) * (float)C_;
    float cls_raw  = (cls_sum + n_unmasked * c0) / fmaxf(cls_pos, 1.0f);
    float cls_loss = (num_pos > 1.0f) ? cls_raw : 0.0f;

    out[0] = heat_loss + box_loss + cls_loss;
}

// ---------------- host launcher ----------------
extern "C" void kernel_launch(void* const* d_in, const int* in_sizes, int n_in,
                              void* d_out, int out_size, void* d_ws, size_t ws_size,
                              hipStream_t stream) {
    const float* pred_heat  = (const float*)d_in[0];   // (B,1,H,W)
    const float* pred_boxes = (const float*)d_in[1];   // (B,4,H,W)
    const float* pred_cls   = (const float*)d_in[2];   // (B,43,H,W)
    const float* bboxes     = (const float*)d_in[3];   // (B,M,4)
    const int*   labels     = (const int*)d_in[4];     // (B,M) int32 (JAX x64 off)
    float*       out        = (float*)d_out;

    char* ws = (char*)d_ws;
    float* acc  = (float*)ws;                          // 8 floats
    float* objx = (float*)(ws + 64);                   // B*M floats (row 0)
    float* objy = objx + (B_ * M_);                    // row stride 1024 floats
    float* objr = objy + (B_ * M_);
    unsigned long long* clsbits =
        (unsigned long long*)(ws + 16384);             // B*H*W u64 (~3.3 MB)

    // stage 0: zero bitmask + accumulators
    {
        int n = B_ * HW_ * 2;                          // 32-bit words
        int grid = (n + 255) / 256;
        afl_zero_kernel<<<grid, 256, 0, stream>>>((unsigned int*)clsbits, acc);
    }
    // stage 1: per-object scatter + box L1
    {
        int n = B_ * M_;
        afl_obj_kernel<<<(n + 255) / 256, 256, 0, stream>>>(
            bboxes, labels, pred_boxes, objx, objy, objr, clsbits, acc);
    }
    // stage 2: fused heatmap/focal/class pass, TDM-staged LDS, WMMA reductions
    {
        int grid = B_ * (HW_ / 256);                   // 1600 blocks
        afl_main_kernel<<<grid, 256, 0, stream>>>(
            pred_heat, pred_cls, objx, clsbits, acc);
    }
    // stage 3: finalize
    afl_final_kernel<<<1, 1, 0, stream>>>(acc, out);
}